// AvgPoolCustom_49314814492846
// MI455X (gfx1250) — compile-verified
//
#include <hip/hip_runtime.h>

// AvgPoolCustom: out[b,c,j,i] = sum_{a,r} x[b,c,a,r] * A[a,j] * R[b,i,r]
// x: (16, 256, 360, 180) fp32.  A: arc j averages a in [90j, 90j+90) with 1/90.
// R: per-batch ring coverage, rows normalized by (bound-lower) analytically.
// Memory-bound: 1.06 GB single-pass stream -> ~46 us floor @ 23.3 TB/s.
// Contraction runs on V_WMMA_F32_16X16X4_F32 (M=16 channels, N=16 rings padded, K=4 r's).

typedef float v2f __attribute__((ext_vector_type(2)));
typedef float v8f __attribute__((ext_vector_type(8)));

#define N_RINGS 180
#define N_ARCS  360

__global__ __launch_bounds__(256) void arcring_pool_kernel(
    const float* __restrict__ features,
    const float* __restrict__ bsize_x,
    float* __restrict__ out)
{
    __shared__ __align__(16) float Rw[16 * N_RINGS]; // padded ring weights (16 x 180)
    __shared__ __align__(16) float red[8 * 256];     // cross-wave reduction buffer

    const int b  = blockIdx.z;          // batch        0..15
    const int j  = blockIdx.y;          // arc index    0..3
    const int c0 = blockIdx.x * 16;     // channel tile 0..255 step 16

    const int tid   = threadIdx.x;
    const int lane  = tid & 31;
    const int wave  = tid >> 5;
    const int m     = lane & 15;        // WMMA row (channel) / B column (ring)
    const int khalf = lane >> 4;        // 0 -> K{0,1}, 1 -> K{2,3}

    // ---- per-batch ring weight matrix, pre-normalized, arc 1/90 folded in ----
    const float inv_mpp = 1.0f / (bsize_x[b] * 0.1f / 2.0f / (float)N_RINGS);
    for (int idx = tid; idx < 16 * N_RINGS; idx += 256) {
        const int i = idx / N_RINGS;
        const int r = idx - i * N_RINGS;
        float w = 0.0f;
        if (i < 5) {
            const float ringmm[5] = {1.f, 2.f, 3.f, 4.f, 6.f};
            float bnd = fminf(fmaxf(ringmm[i] * inv_mpp, 0.f), (float)N_RINGS);
            float low = (i == 0) ? 0.f
                        : fminf(fmaxf(ringmm[i - 1] * inv_mpp, 0.f), (float)N_RINGS);
            float cov = fmaxf(fminf(bnd, (float)r + 1.f) - fmaxf(low, (float)r), 0.f);
            float len = bnd - low;
            w = (len > 0.f) ? cov / len * (1.0f / 90.0f) : 0.0f;
        }
        Rw[idx] = w;
    }
    __syncthreads();

    // ---- stream the arc's angles through WMMA; K reduces over r chunks ----
    v8f acc = {};
    const float* rowbase =
        features + ((size_t)(b * 256 + c0 + m)) * (size_t)(N_ARCS * N_RINGS) + khalf * 2;
    const float* bbase = &Rw[m * N_RINGS + khalf * 2];

    const int a_end = j * 90 + 90;
    for (int a = j * 90 + wave; a < a_end; a += 8) {
        const float* arow = rowbase + (size_t)a * N_RINGS;
        #pragma unroll 5
        for (int rc = 0; rc < 45; ++rc) {
            v2f afrag = *(const v2f*)(arow + rc * 4);    // global_load_b64 stream
            v2f bfrag = *(const v2f*)(bbase + rc * 4);   // ds_load_b64
            acc = __builtin_amdgcn_wmma_f32_16x16x4_f32(
                false, afrag, false, bfrag, (short)0, acc, false, false);
        }
    }

    // ---- cross-wave reduction: 8 partial 16x16 tiles -> one, then scatter ----
    *(v8f*)(&red[wave * 256 + lane * 8]) = acc;
    __syncthreads();

    // thread t owns C element (srcLane = t>>3, vgpr = t&7)
    const int laneSrc = tid >> 3;
    const int v       = tid & 7;
    float s = 0.f;
    #pragma unroll
    for (int w = 0; w < 8; ++w) s += red[w * 256 + tid];

    const int M = v + ((laneSrc >= 16) ? 8 : 0);   // channel within tile
    const int N = laneSrc & 15;                    // ring index (only 0..4 valid)
    if (N < 5) {
        out[(((size_t)(b * 256 + c0 + M)) * 4 + j) * 5 + N] = s;
    }
}

extern "C" void kernel_launch(void* const* d_in, const int* in_sizes, int n_in,
                              void* d_out, int out_size, void* d_ws, size_t ws_size,
                              hipStream_t stream) {
    const float* features = (const float*)d_in[0];  // (16,256,360,180) fp32
    const float* bsize_x  = (const float*)d_in[1];  // (16,) fp32
    float* out = (float*)d_out;                     // (16,256,4,5) fp32

    dim3 grid(16, 4, 16);   // (channel tiles, arcs, batch)
    arcring_pool_kernel<<<grid, dim3(256, 1, 1), 0, stream>>>(features, bsize_x, out);
}